// MatchAssignment_17248588660906
// MI455X (gfx1250) — compile-verified
//
#include <hip/hip_runtime.h>
#include <float.h>

#define NN 8192
#define DD 256
#define EE 262144

typedef float v2f __attribute__((ext_vector_type(2)));
typedef float v4f __attribute__((ext_vector_type(4)));
typedef float v8f __attribute__((ext_vector_type(8)));

#define NEGF (-3.402823466e38f)

// ---------------------------------------------------------------------------
// 1) Fill sim with -FLT_MAX (nontemporal: sim > L2, don't pollute)
// ---------------------------------------------------------------------------
__global__ __launch_bounds__(256) void fill_neg_kernel(v4f* __restrict__ sim4) {
    size_t i = (size_t)blockIdx.x * blockDim.x + threadIdx.x;   // 16M v4f
    v4f v = {NEGF, NEGF, NEGF, NEGF};
    __builtin_nontemporal_store(v, sim4 + i);
}

// ---------------------------------------------------------------------------
// 2) Edge dots via WMMA f32 16x16x4, diagonal extraction, fused scatter.
//    One wave handles 16 edges. A = 16 src rows (16xK), B = 16 dst rows as
//    columns (Kx16); chain over K=256 in steps of 4. diag(D) = edge dots.
//    f32 A-layout: lanes 0-15 hold K=k,k+1 ; lanes 16-31 hold K=k+2,k+3
//    -> one contiguous float2 load per lane per matrix per step.
// ---------------------------------------------------------------------------
__global__ __launch_bounds__(256) void edge_wmma_kernel(
    const float* __restrict__ x, const int* __restrict__ src,
    const int* __restrict__ dst, float* __restrict__ sim) {
    int wave = (int)((blockIdx.x * blockDim.x + threadIdx.x) >> 5);
    int lane = threadIdx.x & 31;
    int e0   = wave * 16;
    if (e0 >= EE) return;

    int m  = lane & 15;            // row-of-A / col-of-B owned by this lane
    int rA = src[e0 + m];
    int rB = dst[e0 + m];
    int koff = (lane >= 16) ? 2 : 0;
    const float* pa = x + (size_t)rA * DD + koff;
    const float* pb = x + (size_t)rB * DD + koff;

    v8f c = {};
#pragma unroll 8
    for (int k = 0; k < DD; k += 4) {
        v2f a = *(const v2f*)(pa + k);
        v2f b = *(const v2f*)(pb + k);
        // D = A*B + C, full fp32
        c = __builtin_amdgcn_wmma_f32_16x16x4_f32(
                false, a, false, b, (short)0, c, false, false);
    }

    // C/D layout: VGPR r, lanes 0-15 -> (M=r, N=lane); lanes 16-31 -> (M=r+8, N=lane-16)
    // diag element t:  t<8 : c[t] @ lane t ;  t>=8 : c[t-8] @ lane t+16
    float cand = 0.0f;
    if (lane < 8)        cand = c[lane];
    else if (lane >= 24) cand = c[lane - 24];
    int srcLane = (m < 8) ? m : (m + 16);
    float dval = __shfl(cand, srcLane, 32);

    if (lane < 16) {
        // scale: xs = x/d^0.25 on both sides -> dot/sqrt(D) = dot/16
        sim[(size_t)rA * NN + (size_t)rB] = dval * 0.0625f;
    }
}

// ---------------------------------------------------------------------------
// Reduction helpers (wave32)
// ---------------------------------------------------------------------------
__device__ inline float waveMax(float v) {
#pragma unroll
    for (int o = 16; o > 0; o >>= 1) v = fmaxf(v, __shfl_down(v, o, 32));
    return v;
}
__device__ inline float waveSum(float v) {
#pragma unroll
    for (int o = 16; o > 0; o >>= 1) v += __shfl_down(v, o, 32);
    return v;
}

// ---------------------------------------------------------------------------
// 3) Row stats: one block per row; row cached in LDS (32KB of 320KB/WGP),
//    single global read, then max + sum(exp) from LDS.
// ---------------------------------------------------------------------------
__global__ __launch_bounds__(256) void row_stats_kernel(
    const float* __restrict__ sim, float* __restrict__ rowmax,
    float* __restrict__ rowinv) {
    __shared__ float srow[NN];
    __shared__ float red[8];
    int tid = threadIdx.x, wid = tid >> 5, lane = tid & 31;

    const v4f* rp = (const v4f*)(sim + (size_t)blockIdx.x * NN);
    v4f* s4 = (v4f*)srow;
    for (int j = tid; j < NN / 4; j += 256) s4[j] = rp[j];
    __syncthreads();

    float m = NEGF;
    for (int j = tid; j < NN; j += 256) m = fmaxf(m, srow[j]);
    m = waveMax(m);
    if (lane == 0) red[wid] = m;
    __syncthreads();
    if (wid == 0) {
        float t = (lane < 8) ? red[lane] : NEGF;
        t = waveMax(t);
        if (lane == 0) red[0] = t;
    }
    __syncthreads();
    float M = red[0];
    __syncthreads();

    float s = 0.0f;
    for (int j = tid; j < NN; j += 256) s += expf(srow[j] - M); // exp underflow -> exact 0
    s = waveSum(s);
    if (lane == 0) red[wid] = s;
    __syncthreads();
    if (tid == 0) {
        float t = 0.0f;
#pragma unroll
        for (int w = 0; w < 8; ++w) t += red[w];
        rowmax[blockIdx.x] = M;
        rowinv[blockIdx.x] = 1.0f / t;
    }
}

// ---------------------------------------------------------------------------
// 4) Column stats: columns across lanes (coalesced), rows tiled across
//    blockIdx.y into deterministic partial buffers, then tiny reductions.
// ---------------------------------------------------------------------------
__global__ __launch_bounds__(256) void col_pmax_kernel(
    const float* __restrict__ sim, float* __restrict__ pmax) {
    int col = blockIdx.x * 256 + threadIdx.x;
    int r0  = blockIdx.y * 256;
    float m = NEGF;
#pragma unroll 4
    for (int r = 0; r < 256; ++r)
        m = fmaxf(m, __builtin_nontemporal_load(&sim[(size_t)(r0 + r) * NN + col]));
    pmax[(size_t)blockIdx.y * NN + col] = m;
}

__global__ __launch_bounds__(256) void col_rmax_kernel(
    const float* __restrict__ pmax, float* __restrict__ colmax) {
    int col = blockIdx.x * 256 + threadIdx.x;
    float m = NEGF;
#pragma unroll
    for (int c = 0; c < 32; ++c) m = fmaxf(m, pmax[(size_t)c * NN + col]);
    colmax[col] = m;
}

__global__ __launch_bounds__(256) void col_psum_kernel(
    const float* __restrict__ sim, const float* __restrict__ colmax,
    float* __restrict__ psum) {
    int col = blockIdx.x * 256 + threadIdx.x;
    int r0  = blockIdx.y * 256;
    float cm = colmax[col];
    float s = 0.0f;
#pragma unroll 4
    for (int r = 0; r < 256; ++r)
        s += expf(__builtin_nontemporal_load(&sim[(size_t)(r0 + r) * NN + col]) - cm);
    psum[(size_t)blockIdx.y * NN + col] = s;
}

__global__ __launch_bounds__(256) void col_rsum_kernel(
    const float* __restrict__ psum, float* __restrict__ colinv) {
    int col = blockIdx.x * 256 + threadIdx.x;
    float s = 0.0f;
#pragma unroll
    for (int c = 0; c < 32; ++c) s += psum[(size_t)c * NN + col];
    colinv[col] = 1.0f / s;
}

// ---------------------------------------------------------------------------
// 5) Fused final: scores = 0.5*(exp(sim-rm)*rinv + exp(sim-cm)*cinv)
//    v4f streaming; row/col stats (128KB) live in L2.
// ---------------------------------------------------------------------------
__global__ __launch_bounds__(256) void final_kernel(
    const float* __restrict__ sim, const float* __restrict__ rowmax,
    const float* __restrict__ rowinv, const float* __restrict__ colmax,
    const float* __restrict__ colinv, float* __restrict__ scores) {
    size_t idx4 = (size_t)blockIdx.x * blockDim.x + threadIdx.x; // 16M v4f
    size_t i  = idx4 >> 11;            // 2048 v4f per row
    size_t j0 = (idx4 & 2047) << 2;

    v4f v = __builtin_nontemporal_load((const v4f*)sim + idx4);
    float rm = rowmax[i], ri = rowinv[i];
    v4f cm = *(const v4f*)(colmax + j0);
    v4f ci = *(const v4f*)(colinv + j0);

    v4f o;
    o.x = 0.5f * (expf(v.x - rm) * ri + expf(v.x - cm.x) * ci.x);
    o.y = 0.5f * (expf(v.y - rm) * ri + expf(v.y - cm.y) * ci.y);
    o.z = 0.5f * (expf(v.z - rm) * ri + expf(v.z - cm.z) * ci.z);
    o.w = 0.5f * (expf(v.w - rm) * ri + expf(v.w - cm.w) * ci.w);
    __builtin_nontemporal_store(o, (v4f*)scores + idx4);
}

// ---------------------------------------------------------------------------
extern "C" void kernel_launch(void* const* d_in, const int* in_sizes, int n_in,
                              void* d_out, int out_size, void* d_ws, size_t ws_size,
                              hipStream_t stream) {
    const float* x  = (const float*)d_in[0];
    const int*   ei = (const int*)d_in[1];   // (2,E): row0=src, row1=dst
    const int*   src = ei;
    const int*   dst = ei + EE;

    float* scores = (float*)d_out;                       // output 0: N*N
    float* sim    = scores + (size_t)NN * NN;            // output 1: N*N

    float* ws     = (float*)d_ws;
    float* rowmax = ws;                 // N
    float* rowinv = ws + NN;            // N
    float* colmax = ws + 2 * NN;        // N
    float* colinv = ws + 3 * NN;        // N
    float* pbuf   = ws + 4 * NN;        // 32*N partials (1 MB)

    const size_t n4 = (size_t)NN * NN / 4;               // 16M v4f

    fill_neg_kernel<<<(unsigned)(n4 / 256), 256, 0, stream>>>((v4f*)sim);

    edge_wmma_kernel<<<(EE / 16) / 8, 256, 0, stream>>>(x, src, dst, sim);

    row_stats_kernel<<<NN, 256, 0, stream>>>(sim, rowmax, rowinv);

    dim3 cgrid(NN / 256, 32);
    col_pmax_kernel<<<cgrid, 256, 0, stream>>>(sim, pbuf);
    col_rmax_kernel<<<NN / 256, 256, 0, stream>>>(pbuf, colmax);
    col_psum_kernel<<<cgrid, 256, 0, stream>>>(sim, colmax, pbuf);
    col_rsum_kernel<<<NN / 256, 256, 0, stream>>>(pbuf, colinv);

    final_kernel<<<(unsigned)(n4 / 256), 256, 0, stream>>>(
        sim, rowmax, rowinv, colmax, colinv, scores);
}